// SwitchGate_27547920236701
// MI455X (gfx1250) — compile-verified
//
#include <hip/hip_runtime.h>
#include <hip/hip_bf16.h>

// SwitchGate for MI455X (gfx1250, wave32).
// out[b,s,e] = softmax_e(X[b,s,:]@W + bias)[e] / (sum_b softmax + eps) * capacity
//
// HBM-bound: ~268 MB traffic -> ~11.5 us floor at 23.3 TB/s. GEMV uses
// V_WMMA_F32_16X16X4_F32 (full f32 precision on the matrix pipe).
// 4-way K-split across waves (32 waves/block) to get enough loads in flight
// per SIMD (Little's law: ~14 KB/SIMD needed at 600ns latency, ~23 B/ns/SIMD).

typedef __attribute__((ext_vector_type(2))) float v2f;
typedef __attribute__((ext_vector_type(8))) float v8f;

#define B_DIM 8
#define S_DIM 4096
#define D_DIM 2048
#define E_DIM 16
#define KSPLIT 4
#define KCHUNK (D_DIM / KSPLIT)   // 512
#define CAPACITY 8.0f
#define EPSILON 1e-6f

__global__ __launch_bounds__(1024)
void switch_gate_wmma_kernel(const float* __restrict__ X,
                             const float* __restrict__ W,
                             const float* __restrict__ bias,
                             float* __restrict__ out) {
  // partial logit tiles: 32 waves x 16 tokens x 16 experts = 32 KB
  __shared__ float part[KSPLIT * B_DIM][16][E_DIM];
  // final gate tiles for the 8 batches of this s-tile: 8 KB
  __shared__ float gate[B_DIM][16][E_DIM];

  const int tid  = threadIdx.x;
  const int ww   = tid >> 5;        // wave 0..31
  const int lane = tid & 31;
  const int half = lane >> 4;       // 0 = lanes 0-15, 1 = lanes 16-31
  const int col  = lane & 15;       // N (expert) for B/C; M row for A
  const int b    = ww & 7;          // batch index
  const int q    = ww >> 3;         // K-slice 0..3
  const int s0   = blockIdx.x * 16; // 16 s-values per block, all 8 batches
  const int k0   = q * KCHUNK;

  // A-matrix 16x4 f32 layout: lanes 0-15 hold K={kk,kk+1}, lanes 16-31 K={kk+2,kk+3};
  // row M = col -> token (b, s0+col).
  const float* __restrict__ xrow =
      X + ((size_t)b * S_DIM + (size_t)(s0 + col)) * D_DIM + k0 + 2 * half;
  // B-matrix 4x16 f32 (K x E), W row-major [D,E]:
  // VGPR0: rows {kk, kk+2}; VGPR1: rows {kk+1, kk+3}; N = col.
  const float* __restrict__ wcol = W + (size_t)(k0 + 2 * half) * E_DIM + col;

  v8f c = {};  // partial logits: lane holds (token j+8*half, expert col), j=0..7
  #pragma unroll 8
  for (int kk = 0; kk < KCHUNK; kk += 4) {
    v2f a = *(const v2f*)(xrow + kk);                 // global_load_b64
    v2f bfrag;
    bfrag.x = wcol[kk * E_DIM];                       // W[k0+kk+2*half  ][col]
    bfrag.y = wcol[kk * E_DIM + E_DIM];               // W[k0+kk+2*half+1][col]
    c = __builtin_amdgcn_wmma_f32_16x16x4_f32(
        /*neg_a=*/false, a, /*neg_b=*/false, bfrag,
        /*c_mod=*/(short)0, c, /*reuse_a=*/false, /*reuse_b=*/false);
  }

  // publish partial tile (C layout: VGPR j -> token j+8*half, expert col)
  #pragma unroll
  for (int j = 0; j < 8; ++j) part[ww][j + 8 * half][col] = c[j];
  __syncthreads();

  // Reduce the 4 K-slices (fixed order -> deterministic), add bias, softmax
  // over experts (16 lanes of this half-wave; xor masks 1,2,4,8 stay inside
  // the 16-lane group). All 32 waves compute redundantly to keep control
  // flow uniform; only q==0 waves publish.
  const float bval = bias[col];
  float g[8];
  #pragma unroll
  for (int j = 0; j < 8; ++j) {
    const int m = j + 8 * half;
    float l = bval + part[b][m][col] + part[b + 8][m][col]
                   + part[b + 16][m][col] + part[b + 24][m][col];
    float mx = l;
    mx = fmaxf(mx, __shfl_xor(mx, 1));
    mx = fmaxf(mx, __shfl_xor(mx, 2));
    mx = fmaxf(mx, __shfl_xor(mx, 4));
    mx = fmaxf(mx, __shfl_xor(mx, 8));
    float e = __expf(l - mx);
    float s = e;
    s += __shfl_xor(s, 1);
    s += __shfl_xor(s, 2);
    s += __shfl_xor(s, 4);
    s += __shfl_xor(s, 8);
    g[j] = e / s;  // gate score; TOPK == E so the top-k mask is all-ones
    if (q == 0) gate[b][j + 8 * half][col] = g[j];
  }
  __syncthreads();

  // denominator = sum over batch (fixed order -> deterministic), scale, store
  if (q == 0) {
    #pragma unroll
    for (int j = 0; j < 8; ++j) {
      const int m = j + 8 * half;
      float denom = EPSILON;
      #pragma unroll
      for (int bb = 0; bb < B_DIM; ++bb) denom += gate[bb][m][col];
      out[((size_t)b * S_DIM + (size_t)(s0 + m)) * E_DIM + col] =
          g[j] / denom * CAPACITY;
    }
  }
}

extern "C" void kernel_launch(void* const* d_in, const int* in_sizes, int n_in,
                              void* d_out, int out_size, void* d_ws, size_t ws_size,
                              hipStream_t stream) {
  const float* X    = (const float*)d_in[0];  // [8,4096,2048] f32
  const float* W    = (const float*)d_in[1];  // [2048,16] f32
  const float* bias = (const float*)d_in[2];  // [16] f32
  float* out        = (float*)d_out;          // [8,4096,16] f32

  dim3 grid(S_DIM / 16);  // 256 blocks; each covers 16 s-values x all 8 batches
  dim3 block(1024);       // 32 waves: 8 batches x 4 K-slices
  switch_gate_wmma_kernel<<<grid, block, 0, stream>>>(X, W, bias, out);
}